// ChannelHyperedgeInteraction_91079076479652
// MI455X (gfx1250) — compile-verified
//
#include <hip/hip_runtime.h>
#include <math.h>

typedef float v2f __attribute__((ext_vector_type(2)));
typedef float v8f __attribute__((ext_vector_type(8)));

#define B_  16
#define C_  128
#define L_  256
#define D_  512

// ---------------------------------------------------------------------------
// K1: masked mean over L (the only HBM-bound phase: streams 1 GiB of x).
// One block per (b,c) row. 256 threads = 128 col-groups (float4) x 2 L-halves.
// ---------------------------------------------------------------------------
__global__ __launch_bounds__(256)
void masked_mean_kernel(const float* __restrict__ x,
                        const float* __restrict__ mask,
                        float* __restrict__ E,
                        float* __restrict__ total)
{
    const int bc    = blockIdx.x;       // 0..B_*C_-1
    const int tid   = threadIdx.x;      // 0..255
    const int colv  = tid & 127;        // float4 column group (4 cols)
    const int lhalf = tid >> 7;         // 0 or 1

    __shared__ float  smask[L_];
    __shared__ float  sred[256];
    __shared__ float4 sacc[128];

    smask[tid] = mask[(size_t)bc * L_ + tid];
    __syncthreads();

    // mask sum (tree reduce in separate array so smask survives)
    sred[tid] = smask[tid];
    __syncthreads();
    for (int s = 128; s > 0; s >>= 1) {
        if (tid < s) sred[tid] += sred[tid + s];
        __syncthreads();
    }
    const float msum = sred[0];
    if (tid == 0) total[bc] = msum;

    const float4* __restrict__ x4 = reinterpret_cast<const float4*>(x);
    const size_t rowbase = (size_t)bc * L_ * (D_ / 4);
    const int l0 = lhalf * 128;

    float4 acc = make_float4(0.f, 0.f, 0.f, 0.f);
    #pragma unroll 4
    for (int l = 0; l < 128; ++l) {
        const size_t idx = rowbase + (size_t)(l0 + l) * (D_ / 4) + colv;
        __builtin_prefetch(&x4[idx + 8 * (D_ / 4)], 0, 0);   // global_prefetch_b8
        const float4 xv = x4[idx];
        const float  m  = smask[l0 + l];
        acc.x += xv.x * m; acc.y += xv.y * m;
        acc.z += xv.z * m; acc.w += xv.w * m;
    }
    if (lhalf == 1) sacc[colv] = acc;
    __syncthreads();
    if (lhalf == 0) {
        const float4 o = sacc[colv];
        const float inv = 1.0f / fmaxf(msum, 1.0f);
        float4 r;
        r.x = (acc.x + o.x) * inv; r.y = (acc.y + o.y) * inv;
        r.z = (acc.z + o.z) * inv; r.w = (acc.w + o.w) * inv;
        reinterpret_cast<float4*>(E)[(size_t)bc * (D_ / 4) + colv] = r;
    }
}

// ---------------------------------------------------------------------------
// K2: overlap modulation factor F[b,i,j] = 0.5 + (mask_i . mask_j)/max(ti+tj,1)
// WMMA f32 16x16x4, one wave per 16x16 tile, K = L_ = 256.
// ---------------------------------------------------------------------------
__global__ __launch_bounds__(32)
void overlap_factor_wmma(const float* __restrict__ mask,
                         const float* __restrict__ total,
                         float* __restrict__ F)
{
    const int b    = blockIdx.z;
    const int i0   = blockIdx.y * 16;
    const int j0   = blockIdx.x * 16;
    const int lane = threadIdx.x;
    const int r16  = lane & 15;
    const int koff = (lane >> 4) * 2;

    const float* __restrict__ mb = mask + (size_t)b * C_ * L_;

    v8f acc = {};
    for (int k = 0; k < L_; k += 4) {
        const float2 av = *reinterpret_cast<const float2*>(mb + (size_t)(i0 + r16) * L_ + k + koff);
        const float2 bv = *reinterpret_cast<const float2*>(mb + (size_t)(j0 + r16) * L_ + k + koff);
        v2f a;  a[0]  = av.x; a[1]  = av.y;   // A: 16x4, lane=M, vgpr0/1 = K koff/koff+1
        v2f bb; bb[0] = bv.x; bb[1] = bv.y;   // B: 4x16 = mask^T, lane=N
        acc = __builtin_amdgcn_wmma_f32_16x16x4_f32(false, a, false, bb,
                                                    (short)0, acc, false, false);
    }
    const float* tb = total + (size_t)b * C_;
    #pragma unroll
    for (int r = 0; r < 8; ++r) {
        const int i = i0 + ((lane >> 4) * 8) + r;
        const int j = j0 + r16;
        const float tp = tb[i] + tb[j];
        F[((size_t)b * C_ + i) * C_ + j] = 0.5f + acc[r] / fmaxf(tp, 1.0f);
    }
}

// ---------------------------------------------------------------------------
// K3: generic batched NN GEMM  C = act(A(MxK) @ B(KxN) + bias)
// A,B,C row-major; one wave per 16x16 tile; act: 0=none, 1=exact GELU.
// Data is L2-resident so direct global loads are sufficient.
// ---------------------------------------------------------------------------
__global__ __launch_bounds__(32)
void gemm_nn_wmma(const float* __restrict__ A, int lda, long sA,
                  const float* __restrict__ Bm, int ldb, long sB,
                  const float* __restrict__ bias,
                  float* __restrict__ Cm, int ldc, long sC,
                  int K, int act)
{
    const int bz   = blockIdx.z;
    const int n0   = blockIdx.x * 16;
    const int m0   = blockIdx.y * 16;
    const int lane = threadIdx.x;
    const int r16  = lane & 15;
    const int koff = (lane >> 4) * 2;

    const float* __restrict__ Ab = A  + (size_t)bz * sA;
    const float* __restrict__ Bb = Bm + (size_t)bz * sB;
    float*       __restrict__ Cb = Cm + (size_t)bz * sC;

    v8f acc = {};
    for (int k = 0; k < K; k += 4) {
        const float2 av = *reinterpret_cast<const float2*>(Ab + (size_t)(m0 + r16) * lda + k + koff);
        v2f a;  a[0]  = av.x; a[1]  = av.y;
        v2f bb; bb[0] = Bb[(size_t)(k + koff)     * ldb + n0 + r16];
                bb[1] = Bb[(size_t)(k + koff + 1) * ldb + n0 + r16];
        acc = __builtin_amdgcn_wmma_f32_16x16x4_f32(false, a, false, bb,
                                                    (short)0, acc, false, false);
    }
    #pragma unroll
    for (int r = 0; r < 8; ++r) {
        const int m = m0 + ((lane >> 4) * 8) + r;
        const int n = n0 + r16;
        float v = acc[r];
        if (bias) v += bias[n];
        if (act == 1) v = 0.5f * v * (1.0f + erff(v * 0.70710678118654752f)); // exact GELU
        Cb[(size_t)m * ldc + n] = v;
    }
}

// ---------------------------------------------------------------------------
// K4: attention scores  S[b,i,j] = (q_i . k_j) / sqrt(d) * F[b,i,j]
// NT GEMM over qkv (q cols 0..511, k cols 512..1023), K = D_ = 512.
// ---------------------------------------------------------------------------
__global__ __launch_bounds__(32)
void scores_wmma(const float* __restrict__ qkv,
                 const float* __restrict__ F,
                 float* __restrict__ S)
{
    const int b    = blockIdx.z;
    const int i0   = blockIdx.y * 16;
    const int j0   = blockIdx.x * 16;
    const int lane = threadIdx.x;
    const int r16  = lane & 15;
    const int koff = (lane >> 4) * 2;

    const float* __restrict__ qb = qkv + (size_t)b * C_ * (3 * D_);

    v8f acc = {};
    for (int k = 0; k < D_; k += 4) {
        const float2 av = *reinterpret_cast<const float2*>(qb + (size_t)(i0 + r16) * (3 * D_) + k + koff);
        const float2 bv = *reinterpret_cast<const float2*>(qb + (size_t)(j0 + r16) * (3 * D_) + D_ + k + koff);
        v2f a;  a[0]  = av.x; a[1]  = av.y;
        v2f bb; bb[0] = bv.x; bb[1] = bv.y;
        acc = __builtin_amdgcn_wmma_f32_16x16x4_f32(false, a, false, bb,
                                                    (short)0, acc, false, false);
    }
    const float scale = 0.04419417382415922f; // 1/sqrt(512)
    #pragma unroll
    for (int r = 0; r < 8; ++r) {
        const int i = i0 + ((lane >> 4) * 8) + r;
        const int j = j0 + r16;
        const size_t idx = ((size_t)b * C_ + i) * C_ + j;
        S[idx] = acc[r] * scale * F[idx];
    }
}

// ---------------------------------------------------------------------------
// K5: row softmax over 128 entries, in place.
// ---------------------------------------------------------------------------
__global__ __launch_bounds__(128)
void softmax128_kernel(float* __restrict__ S)
{
    const int row = blockIdx.x;   // B_*C_ rows
    const int tid = threadIdx.x;  // 128
    __shared__ float sr[128];

    const size_t base = (size_t)row * C_;
    const float v = S[base + tid];

    sr[tid] = v; __syncthreads();
    for (int s = 64; s > 0; s >>= 1) {
        if (tid < s) sr[tid] = fmaxf(sr[tid], sr[tid + s]);
        __syncthreads();
    }
    const float mx = sr[0];
    __syncthreads();

    const float e = expf(v - mx);
    sr[tid] = e; __syncthreads();
    for (int s = 64; s > 0; s >>= 1) {
        if (tid < s) sr[tid] += sr[tid + s];
        __syncthreads();
    }
    S[base + tid] = e / sr[0];
}

// ---------------------------------------------------------------------------
// K6: Out = LayerNorm(X + Y) * g + be   (row length D_ = 512, 256 threads)
// ---------------------------------------------------------------------------
__global__ __launch_bounds__(256)
void add_ln_kernel(const float* __restrict__ X,
                   const float* __restrict__ Y,
                   const float* __restrict__ g,
                   const float* __restrict__ be,
                   float* __restrict__ Out)
{
    const int row = blockIdx.x;   // B_*C_
    const int tid = threadIdx.x;  // 256
    __shared__ float s1[256], s2[256];

    const size_t base = (size_t)row * D_;
    const int c0 = tid * 2;
    float vx = X[base + c0]     + Y[base + c0];
    float vy = X[base + c0 + 1] + Y[base + c0 + 1];

    s1[tid] = vx + vy;
    s2[tid] = vx * vx + vy * vy;
    __syncthreads();
    for (int s = 128; s > 0; s >>= 1) {
        if (tid < s) { s1[tid] += s1[tid + s]; s2[tid] += s2[tid + s]; }
        __syncthreads();
    }
    const float mean = s1[0] * (1.0f / D_);
    const float var  = s2[0] * (1.0f / D_) - mean * mean;
    const float rstd = rsqrtf(var + 1e-5f);

    Out[base + c0]     = (vx - mean) * rstd * g[c0]     + be[c0];
    Out[base + c0 + 1] = (vy - mean) * rstd * g[c0 + 1] + be[c0 + 1];
}

// ---------------------------------------------------------------------------
// Host launcher
// ---------------------------------------------------------------------------
extern "C" void kernel_launch(void* const* d_in, const int* in_sizes, int n_in,
                              void* d_out, int out_size, void* d_ws, size_t ws_size,
                              hipStream_t stream) {
    const float* x    = (const float*)d_in[0];
    const float* mask = (const float*)d_in[1];
    const float* Wqkv = (const float*)d_in[2];
    const float* bqkv = (const float*)d_in[3];
    const float* Wo   = (const float*)d_in[4];
    const float* bo   = (const float*)d_in[5];
    const float* W1   = (const float*)d_in[6];
    const float* b1   = (const float*)d_in[7];
    const float* W2   = (const float*)d_in[8];
    const float* b2   = (const float*)d_in[9];
    const float* g1   = (const float*)d_in[10];
    const float* be1  = (const float*)d_in[11];
    const float* g2   = (const float*)d_in[12];
    const float* be2  = (const float*)d_in[13];
    float* out = (float*)d_out;

    // workspace layout (floats)
    float* ws = (float*)d_ws;
    const size_t MN = (size_t)B_ * C_;          // 2048 rows
    float* E       = ws;                        // 2048*512
    float* total   = E       + MN * D_;         // 2048
    float* F       = total   + MN;              // 16*128*128
    float* qkv     = F       + (size_t)B_ * C_ * C_;   // 2048*1536
    float* S       = qkv     + MN * 3 * D_;     // 16*128*128
    float* attnout = S       + (size_t)B_ * C_ * C_;   // 2048*512
    float* proj    = attnout + MN * D_;         // 2048*512
    float* E1      = proj    + MN * D_;         // 2048*512
    float* h       = E1      + MN * D_;         // 2048*1024
    float* out2    = h       + MN * 2 * D_;     // 2048*512

    // 1) masked mean + mask totals (streams 1 GiB of x, HBM-bound)
    masked_mean_kernel<<<dim3(MN), dim3(256), 0, stream>>>(x, mask, E, total);

    // 2) overlap modulation factor (WMMA, K=256)
    overlap_factor_wmma<<<dim3(C_ / 16, C_ / 16, B_), dim3(32), 0, stream>>>(mask, total, F);

    // 3) qkv = E @ Wqkv + bqkv   (2048x512 @ 512x1536)
    gemm_nn_wmma<<<dim3((3 * D_) / 16, MN / 16, 1), dim3(32), 0, stream>>>(
        E, D_, 0, Wqkv, 3 * D_, 0, bqkv, qkv, 3 * D_, 0, D_, 0);

    // 4) S = (q k^T)/sqrt(d) * F   (per batch 128x128, K=512)
    scores_wmma<<<dim3(C_ / 16, C_ / 16, B_), dim3(32), 0, stream>>>(qkv, F, S);

    // 5) softmax rows
    softmax128_kernel<<<dim3(MN), dim3(128), 0, stream>>>(S);

    // 6) attnout = A @ v   (batched: 128x128 @ 128x512 per batch)
    gemm_nn_wmma<<<dim3(D_ / 16, C_ / 16, B_), dim3(32), 0, stream>>>(
        S, C_, (long)C_ * C_,
        qkv + 2 * D_, 3 * D_, (long)C_ * 3 * D_,
        nullptr, attnout, D_, (long)C_ * D_, C_, 0);

    // 7) proj = attnout @ Wo + bo
    gemm_nn_wmma<<<dim3(D_ / 16, MN / 16, 1), dim3(32), 0, stream>>>(
        attnout, D_, 0, Wo, D_, 0, bo, proj, D_, 0, D_, 0);

    // 8) E1 = LN(E + proj)
    add_ln_kernel<<<dim3(MN), dim3(256), 0, stream>>>(E, proj, g1, be1, E1);

    // 9) h = gelu(E1 @ W1 + b1)   (2048x512 @ 512x1024)
    gemm_nn_wmma<<<dim3((2 * D_) / 16, MN / 16, 1), dim3(32), 0, stream>>>(
        E1, D_, 0, W1, 2 * D_, 0, b1, h, 2 * D_, 0, D_, 1);

    // 10) out2 = h @ W2 + b2   (2048x1024 @ 1024x512)
    gemm_nn_wmma<<<dim3(D_ / 16, MN / 16, 1), dim3(32), 0, stream>>>(
        h, 2 * D_, 0, W2, D_, 0, b2, out2, D_, 0, 2 * D_, 0);

    // 11) out = LN(E1 + out2)
    add_ln_kernel<<<dim3(MN), dim3(256), 0, stream>>>(E1, out2, g2, be2, out);

    (void)in_sizes; (void)n_in; (void)out_size; (void)ws_size;
}